// Model_39676907887730
// MI455X (gfx1250) — compile-verified
//
#include <hip/hip_runtime.h>

typedef __attribute__((ext_vector_type(2))) float v2f;
typedef __attribute__((ext_vector_type(8))) float v8f;

#define TN   2048
#define BM   128
#define BN   128
#define BK   32
#define KP   (BK / 2)      // float2 K-pairs per tile
#define LDA2 (BM + 16)     // float2 row stride: 288 dwords == 32 mod 64 banks
#define LDB2 (BN + 16)

// C = A @ B (+ addend), fp32, via V_WMMA_F32_16X16X4_F32.
// 256 threads = 8 waves; wave grid 2(M) x 4(N); each wave: 64x32 = 4x2 WMMA tiles.
// LDS holds K-paired float2 so every fragment load is one aligned ds_load_b64.
// Register-prefetch double buffering overlaps next-tile global loads with WMMA.
__global__ __launch_bounds__(256) void wmma_gemm_f32(
    const float* __restrict__ Ag, const float* __restrict__ Bg,
    const float* __restrict__ addend, float* __restrict__ Cg)
{
    __shared__ v2f As2[KP][LDA2];   // As2[p][m] = (A[m][2p], A[m][2p+1])
    __shared__ v2f Bs2[KP][LDB2];   // Bs2[p][n] = (B[2p][n], B[2p+1][n])

    const int tid  = threadIdx.x;
    const int lane = tid & 31;
    const int wave = tid >> 5;          // 0..7
    const int wm   = wave & 1;          // wave row (x64)
    const int wn   = wave >> 1;         // wave col (x32)
    const int half = lane >> 4;         // 0: lanes 0-15, 1: lanes 16-31
    const int l    = lane & 15;

    const int bm0 = blockIdx.y * BM;
    const int bn0 = blockIdx.x * BN;

    v2f   ra[8];                        // A-tile staging regs (8 pairs/thread)
    float rb0[8], rb1[8];               // B-tile staging regs

    auto load_regs = [&](int k0) {
        // A: 128 rows x 16 pairs = 2048 pairs; 8/thread; 8B vector reads along rows
        #pragma unroll
        for (int i = 0; i < 8; ++i) {
            int idx = tid + i * 256;
            int p   = idx & (KP - 1);    // 0..15
            int m   = idx >> 4;          // 0..127
            ra[i] = *(const v2f*)&Ag[(size_t)(bm0 + m) * TN + (k0 + 2 * p)];
        }
        // B: 16 pair-rows x 128 cols; two coalesced b32 rows per pair
        #pragma unroll
        for (int i = 0; i < 8; ++i) {
            int idx = tid + i * 256;
            int n   = idx & (BN - 1);    // 0..127
            int p   = idx >> 7;          // 0..15
            rb0[i] = Bg[(size_t)(k0 + 2 * p + 0) * TN + (bn0 + n)];
            rb1[i] = Bg[(size_t)(k0 + 2 * p + 1) * TN + (bn0 + n)];
        }
    };
    auto store_lds = [&]() {
        #pragma unroll
        for (int i = 0; i < 8; ++i) {
            int idx = tid + i * 256;
            As2[idx & (KP - 1)][idx >> 4] = ra[i];
        }
        #pragma unroll
        for (int i = 0; i < 8; ++i) {
            int idx = tid + i * 256;
            v2f b; b.x = rb0[i]; b.y = rb1[i];
            Bs2[idx >> 7][idx & (BN - 1)] = b;
        }
    };

    v8f acc[4][2];
    #pragma unroll
    for (int mt = 0; mt < 4; ++mt)
        #pragma unroll
        for (int nt = 0; nt < 2; ++nt) {
            v8f z = {0.f,0.f,0.f,0.f,0.f,0.f,0.f,0.f};
            acc[mt][nt] = z;
        }

    load_regs(0);
    store_lds();
    __syncthreads();

    for (int k0 = 0; k0 < TN; k0 += BK) {
        const bool notlast = (k0 + BK) < TN;
        if (notlast) load_regs(k0 + BK);   // overlap next-tile global loads with WMMA

        #pragma unroll
        for (int kk = 0; kk < BK; kk += 4) {
            const int pr = (kk >> 1) + half;   // pair row: K={kk,kk+1} or {kk+2,kk+3}
            v2f afr[4], bfr[2];
            #pragma unroll
            for (int mt = 0; mt < 4; ++mt)
                afr[mt] = As2[pr][wm * 64 + mt * 16 + l];   // one ds_load_b64
            #pragma unroll
            for (int nt = 0; nt < 2; ++nt)
                bfr[nt] = Bs2[pr][wn * 32 + nt * 16 + l];   // one ds_load_b64
            #pragma unroll
            for (int mt = 0; mt < 4; ++mt)
                #pragma unroll
                for (int nt = 0; nt < 2; ++nt)
                    acc[mt][nt] = __builtin_amdgcn_wmma_f32_16x16x4_f32(
                        false, afr[mt], false, bfr[nt],
                        (short)0, acc[mt][nt], false, false);
        }
        __syncthreads();
        if (notlast) {
            store_lds();
            __syncthreads();
        }
    }

    // epilogue: C/D layout: VGPR r -> row r (lanes 0-15) / r+8 (lanes 16-31), col = lane&15
    const bool hasAdd = (addend != nullptr);
    #pragma unroll
    for (int mt = 0; mt < 4; ++mt) {
        #pragma unroll
        for (int nt = 0; nt < 2; ++nt) {
            #pragma unroll
            for (int r = 0; r < 8; ++r) {
                int row = bm0 + wm * 64 + mt * 16 + r + 8 * half;
                int col = bn0 + wn * 32 + nt * 16 + l;
                size_t o = (size_t)row * TN + col;
                float v = acc[mt][nt][r];
                if (hasAdd) v += addend[o];
                Cg[o] = v;
            }
        }
    }
}

__global__ __launch_bounds__(256) void ew_add_kernel(float* __restrict__ dst,
                                                     const float* __restrict__ src, int n)
{
    int i = blockIdx.x * blockDim.x + threadIdx.x;
    if (i < n) dst[i] += src[i];
}

// out currently holds t5; out = A - (t2 + t3*t4/t5)
__global__ __launch_bounds__(256) void final_kernel(const float* __restrict__ A,
                                                    const float* __restrict__ t2,
                                                    const float* __restrict__ t3,
                                                    const float* __restrict__ t4,
                                                    float* __restrict__ out, int n)
{
    int i = blockIdx.x * blockDim.x + threadIdx.x;
    if (i < n) {
        float t5 = out[i];
        out[i] = A[i] - (t2[i] + t3[i] * t4[i] / t5);
    }
}

extern "C" void kernel_launch(void* const* d_in, const int* in_sizes, int n_in,
                              void* d_out, int out_size, void* d_ws, size_t ws_size,
                              hipStream_t stream)
{
    const float* X = (const float*)d_in[0];
    const float* Y = (const float*)d_in[1];
    float* out = (float*)d_out;

    const size_t MAT = (size_t)TN * TN;
    float* W    = (float*)d_ws;
    float* bufA = W + 0 * MAT;   // A = X@X             (lives to end)
    float* bufB = W + 1 * MAT;   // B = X@Y, later t4
    float* bufC = W + 2 * MAT;   // C = Y@B, later DB/E
    float* bufD = W + 3 * MAT;   // D = Y@A
    float* t2   = W + 4 * MAT;
    float* t3   = W + 5 * MAT;
    // d_out doubles as scratch for CA, then t5.

    dim3 grid(TN / BN, TN / BM);
    dim3 block(256);
    const int n = (int)MAT;
    dim3 egrid((n + 255) / 256);

    // base products
    wmma_gemm_f32<<<grid, block, 0, stream>>>(X, X, nullptr, bufA);       // A = X@X
    wmma_gemm_f32<<<grid, block, 0, stream>>>(X, Y, nullptr, bufB);       // B = X@Y
    wmma_gemm_f32<<<grid, block, 0, stream>>>(Y, bufB, nullptr, bufC);    // C = Y@B
    wmma_gemm_f32<<<grid, block, 0, stream>>>(Y, bufA, nullptr, bufD);    // D = Y@A
    // t2 = A@Y + B
    wmma_gemm_f32<<<grid, block, 0, stream>>>(bufA, Y, bufB, t2);
    // CA = C@A  (into d_out scratch)
    wmma_gemm_f32<<<grid, block, 0, stream>>>(bufC, bufA, nullptr, out);
    // t3 = B@A + C   (C dead after)
    wmma_gemm_f32<<<grid, block, 0, stream>>>(bufB, bufA, bufC, t3);
    // DB = D@B  (reuse C slot)
    wmma_gemm_f32<<<grid, block, 0, stream>>>(bufD, bufB, nullptr, bufC);
    // t4 = D@Y + CA  (reuse B slot)
    wmma_gemm_f32<<<grid, block, 0, stream>>>(bufD, Y, out, bufB);
    // E = DB + CA  (in place on bufC)
    ew_add_kernel<<<egrid, block, 0, stream>>>(bufC, out, n);
    // t5 = E@A  (into d_out, overwrites CA)
    wmma_gemm_f32<<<grid, block, 0, stream>>>(bufC, bufA, nullptr, out);
    // out = A - (t2 + t3*t4/t5)
    final_kernel<<<egrid, block, 0, stream>>>(bufA, t2, t3, bufB, out, n);
}